// SAGE_36197984370866
// MI455X (gfx1250) — compile-verified
//
#include <hip/hip_runtime.h>
#include <math.h>

#define NN 100000
#define NE 1600000
#define DH 128
#define NC 40
#define A_STRIDE 260   // 260 % 64 == 4 -> conflict-free 16-lane column reads

typedef __attribute__((ext_vector_type(2))) float v2f;
typedef __attribute__((ext_vector_type(8))) float v8f;

__global__ void zero_f32v4(float4* __restrict__ p, int n4) {
    int i = blockIdx.x * blockDim.x + threadIdx.x;
    if (i < n4) p[i] = make_float4(0.f, 0.f, 0.f, 0.f);
}

__global__ void deg_kernel(const int* __restrict__ dst, float* __restrict__ deg) {
    int e = blockIdx.x * blockDim.x + threadIdx.x;
    if (e < NE) atomicAdd(&deg[dst[e]], 1.0f);
}

__global__ void deginv_kernel(float* __restrict__ deg) {
    int i = blockIdx.x * blockDim.x + threadIdx.x;
    if (i < NN) deg[i] = 1.0f / fmaxf(deg[i], 1.0f);
}

// One wave per edge. src/dst are wave-uniform -> force scalar with
// readfirstlane so loads/atomics use the SADDR (scalar base + lane offset)
// form: one global_load_b128 (512B/wave coalesced row gather) and 4
// L2-resident global_atomic_add_f32 per lane to the contiguous dst row.
__global__ void scatter_kernel(const float* __restrict__ x,
                               const int* __restrict__ src,
                               const int* __restrict__ dst,
                               float* __restrict__ aggr) {
    const int lane = threadIdx.x & 31;
    const int wave = threadIdx.x >> 5;
    const int e = blockIdx.x * (blockDim.x >> 5) + wave;
    if (e >= NE) return;
    const int s = __builtin_amdgcn_readfirstlane(src[e]);
    const int d = __builtin_amdgcn_readfirstlane(dst[e]);
    float4 v = ((const float4*)(x + (size_t)s * DH))[lane];
    float* o = aggr + (size_t)d * DH + lane * 4;
    atomicAdd(o + 0, v.x);
    atomicAdd(o + 1, v.y);
    atomicAdd(o + 2, v.z);
    atomicAdd(o + 3, v.w);
}

// Fused SAGE layer GEMM:  out = relu?( [aggr*deginv | x] @ [Wl;Wr] + bias )
// Compile-time DIN/DOUT -> B loads/stores become immediate-offset vmem ops.
// One block = 16 node rows; one wave = one 16x16 output tile via
// V_WMMA_F32_16X16X4_F32; K-loop split into Wl half then Wr half.
template <int DIN, int DOUT, bool RELU>
__global__ void sage_gemm(const float* __restrict__ aggr,
                          const float* __restrict__ x,
                          const float* __restrict__ deginv,
                          const float* __restrict__ Wl,   // [DIN][DOUT] row-major
                          const float* __restrict__ Wr,   // [DIN][DOUT] row-major
                          const float* __restrict__ bias, // [DOUT]
                          float* __restrict__ out) {      // [N][DOUT]
    __shared__ float Alds[16 * A_STRIDE];

    const int m0   = blockIdx.x * 16;
    const int tid  = threadIdx.x;
    const int lane = tid & 31;
    const int wave = tid >> 5;
    constexpr int twoK = 2 * DIN;

    // Stage A tile: [aggr * deg_inv | x], 16 rows x 2*DIN cols, into LDS.
    for (int idx = tid; idx < 16 * twoK; idx += blockDim.x) {
        int row = idx / twoK;
        int kk  = idx - row * twoK;
        int g   = m0 + row;
        float v;
        if (kk < DIN) v = aggr[(size_t)g * DIN + kk] * deginv[g];
        else          v = x[(size_t)g * DIN + (kk - DIN)];
        Alds[row * A_STRIDE + kk] = v;
    }
    __syncthreads();

    const int half = lane >> 4;     // 0: K pair (k,k+1); 1: K pair (k+2,k+3)
    const int l15  = lane & 15;
    const int n0   = wave * 16;
    const int col  = n0 + l15;
    const int ccol = (col < DOUT) ? col : 0;  // in-bounds loads, mask at store

    // Per-lane constant bases; loop offsets are compile-time immediates.
    const float* __restrict__ Wlp = Wl + 2 * half * DOUT + ccol;
    const float* __restrict__ Wrp = Wr + 2 * half * DOUT + ccol;
    const float* __restrict__ Ap  = Alds + l15 * A_STRIDE + 2 * half;

    v8f acc = {};
    #pragma unroll 8
    for (int k = 0; k < DIN; k += 4) {          // aggr half  (B from Wl)
        v2f a, b;
        a.x = Ap[k];
        a.y = Ap[k + 1];
        b.x = Wlp[k * DOUT];
        b.y = Wlp[k * DOUT + DOUT];
        acc = __builtin_amdgcn_wmma_f32_16x16x4_f32(
            false, a, false, b, (short)0, acc, false, false);
    }
    #pragma unroll 8
    for (int k = 0; k < DIN; k += 4) {          // self half  (B from Wr)
        v2f a, b;
        a.x = Ap[DIN + k];
        a.y = Ap[DIN + k + 1];
        b.x = Wrp[k * DOUT];
        b.y = Wrp[k * DOUT + DOUT];
        acc = __builtin_amdgcn_wmma_f32_16x16x4_f32(
            false, a, false, b, (short)0, acc, false, false);
    }

    if (col < DOUT) {
        const float bv = bias[col];
        float* __restrict__ op = out + (size_t)(m0 + 8 * half) * DOUT + col;
        #pragma unroll
        for (int r = 0; r < 8; ++r) {           // C/D: lanes 16-31 = rows M+8
            float v = acc[r] + bv;
            if (RELU) v = fmaxf(v, 0.0f);
            op[r * DOUT] = v;
        }
    }
}

__global__ void logsoftmax_kernel(float* __restrict__ h) {
    int i = blockIdx.x * blockDim.x + threadIdx.x;
    if (i >= NN) return;
    float* row = h + (size_t)i * NC;
    float mx = -INFINITY;
    #pragma unroll
    for (int c = 0; c < NC; ++c) mx = fmaxf(mx, row[c]);
    float s = 0.0f;
    #pragma unroll
    for (int c = 0; c < NC; ++c) s += expf(row[c] - mx);
    float lse = mx + logf(s);
    #pragma unroll
    for (int c = 0; c < NC; ++c) row[c] = row[c] - lse;
}

extern "C" void kernel_launch(void* const* d_in, const int* in_sizes, int n_in,
                              void* d_out, int out_size, void* d_ws, size_t ws_size,
                              hipStream_t stream) {
    const float* x   = (const float*)d_in[0];
    const int*   ei  = (const int*)d_in[1];     // [2][NE]
    const int*   src = ei;
    const int*   dst = ei + NE;
    const float* Wl0 = (const float*)d_in[2];
    const float* Wr0 = (const float*)d_in[3];
    const float* b0  = (const float*)d_in[4];
    const float* Wl1 = (const float*)d_in[5];
    const float* Wr1 = (const float*)d_in[6];
    const float* b1  = (const float*)d_in[7];
    const float* Wl2 = (const float*)d_in[8];
    const float* Wr2 = (const float*)d_in[9];
    const float* b2  = (const float*)d_in[10];
    float* hout = (float*)d_out;                 // [NN][NC]

    char* ws = (char*)d_ws;
    size_t off = 0;
    auto carve = [&](size_t bytes) {
        void* p = ws + off;
        off = (off + bytes + 255) & ~(size_t)255;
        return p;
    };
    float* deg  = (float*)carve((size_t)NN * 4);            // becomes deg_inv
    float* aggr = (float*)carve((size_t)NN * DH * 4);
    float* h0   = (float*)carve((size_t)NN * DH * 4);
    float* h1   = (float*)carve((size_t)NN * DH * 4);

    const int ZB = 256;
    dim3 bZ(ZB);
    dim3 gN((NN + ZB - 1) / ZB);
    dim3 gZdeg((NN / 4 + ZB - 1) / ZB);
    dim3 gZaggr((NN * DH / 4 + ZB - 1) / ZB);
    dim3 gE((NE + ZB - 1) / ZB);
    dim3 gScat(NE / 8);                          // 8 waves/block, 1 edge/wave
    dim3 gGemm(NN / 16);                         // 100000 / 16 = 6250

    // degrees -> deg_inv
    zero_f32v4<<<gZdeg, bZ, 0, stream>>>((float4*)deg, NN / 4);
    deg_kernel<<<gE, bZ, 0, stream>>>(dst, deg);
    deginv_kernel<<<gN, bZ, 0, stream>>>(deg);

    // layer 0: x -> h0 (relu)
    zero_f32v4<<<gZaggr, bZ, 0, stream>>>((float4*)aggr, NN * DH / 4);
    scatter_kernel<<<gScat, bZ, 0, stream>>>(x, src, dst, aggr);
    sage_gemm<DH, DH, true><<<gGemm, dim3(256), 0, stream>>>(aggr, x, deg, Wl0, Wr0, b0, h0);

    // layer 1: h0 -> h1 (relu)
    zero_f32v4<<<gZaggr, bZ, 0, stream>>>((float4*)aggr, NN * DH / 4);
    scatter_kernel<<<gScat, bZ, 0, stream>>>(h0, src, dst, aggr);
    sage_gemm<DH, DH, true><<<gGemm, dim3(256), 0, stream>>>(aggr, h0, deg, Wl1, Wr1, b1, h1);

    // layer 2: h1 -> out (no relu), DOUT = 40 -> 3 waves/block
    zero_f32v4<<<gZaggr, bZ, 0, stream>>>((float4*)aggr, NN * DH / 4);
    scatter_kernel<<<gScat, bZ, 0, stream>>>(h1, src, dst, aggr);
    sage_gemm<DH, NC, false><<<gGemm, dim3(96), 0, stream>>>(aggr, h1, deg, Wl2, Wr2, b2, hout);

    // log_softmax in-place on d_out
    logsoftmax_kernel<<<gN, bZ, 0, stream>>>(hout);
}